// chamfer_dist_pc_67577015435957
// MI455X (gfx1250) — compile-verified
//
#include <hip/hip_runtime.h>

typedef __attribute__((ext_vector_type(2))) float v2f;
typedef __attribute__((ext_vector_type(8))) float v8f;

#define PAD_VAL 1000.0f
#define CSPLIT  8   // column-chunk parallelism per row strip

// Single-instruction f32 mins: avoid clang's IEEE canonicalize chains.
__device__ __forceinline__ float min_raw(float a, float b) {
  float r;
  asm("v_min_num_f32 %0, %1, %2" : "=v"(r) : "v"(a), "v"(b));
  return r;
}
// 3-input min: folds two tile results into the accumulator in ONE VALU op.
__device__ __forceinline__ float min3_raw(float a, float b, float c) {
  float r;
  asm("v_min3_num_f32 %0, %1, %2, %3" : "=v"(r) : "v"(a), "v"(b), "v"(c));
  return r;
}

// ---------------------------------------------------------------------------
// Prep: pack each 3-float point into float4 {-2x, -2y, -2z, |p|^2}.
// B-operand form; the A form [p, 1] is recovered with a -0.5 scale once per
// wave. Hot-loop operand fetch becomes a single global_load_b128.
// ---------------------------------------------------------------------------
__global__ __launch_bounds__(256) void pack_points(
    const float* __restrict__ P, float4* __restrict__ W, int total)
{
  const int i = blockIdx.x * 256 + threadIdx.x;
  if (i < total) {
    const float x = P[3 * i + 0];
    const float y = P[3 * i + 1];
    const float z = P[3 * i + 2];
    W[i] = make_float4(-2.0f * x, -2.0f * y, -2.0f * z, x * x + y * y + z * z);
  }
}

// ---------------------------------------------------------------------------
// Partial nearest-neighbor min via V_WMMA_F32_16X16X4_F32:
//   A row m  = [px, py, pz, 1]            (16x4, K=4)
//   B col n  = [-2qx, -2qy, -2qz, |q|^2]  (4x16)
//   C[m][n]  = |p_m|^2                    (ROW-broadcast, loop-invariant)
//   D        = full 16x16 tile of squared distances
//
// VGPR layouts (ISA 7.12.2, wave32):
//   A 16x4 f32 : 2 regs; lanes0-15 -> K0,K1 ; lanes16-31 -> K2,K3 ; M = lane&15
//   B 4x16 f32 : 2 regs; lanes0-15 -> rows K0,K1 ; lanes16-31 -> K2,K3 ; N = lane&15
//   C/D 16x16  : 8 regs; lanes0-15 -> rows 0..7 ; lanes16-31 -> rows 8..15
//
// blockIdx.y selects direction (0: X rows vs Y cols, 1: Y rows vs X cols).
// Each wave owns 64 rows (4 strips) x one column chunk of NC/CSPLIT and
// sweeps it two 16-column tiles at a time (8 WMMAs + 32 v_min3 per step),
// writing raw (unclamped) partial row-mins per chunk.
// ---------------------------------------------------------------------------
__global__ __launch_bounds__(256) void nn_min_wmma(
    const float4* __restrict__ XP, const float4* __restrict__ YP,
    float* __restrict__ d1part, float* __restrict__ d2part, int N, int M)
{
  const int dir  = blockIdx.y;
  const int lane = threadIdx.x & 31;
  const int wave = threadIdx.x >> 5;

  const float4* RP = dir ? YP : XP;            // row points
  const float4* CP = dir ? XP : YP;            // column points
  float* dpart     = dir ? d2part : d1part;
  const int NR = dir ? M : N;                  // #rows
  const int NC = dir ? N : M;                  // #cols

  const int stripsPerBatch = NR / 64;
  const int tasksPerBatch  = stripsPerBatch * CSPLIT;
  const int task    = blockIdx.x * 8 + wave;
  const int batch   = task / tasksPerBatch;
  const int rem     = task % tasksPerBatch;
  const int rowBase = (rem / CSPLIT) * 64;
  const int cs      = rem % CSPLIT;
  const int ccBase  = cs * (NC / CSPLIT);      // first column of this chunk
  const int T       = (NC / CSPLIT) / 16;      // column tiles in this chunk (even)

  const int n16 = lane & 15;
  const int hi  = lane >> 4;

  const float4* Rb = RP + (size_t)batch * NR;
  const float4* Cb = CP + (size_t)batch * NC;

  // Prologue: A strips [p, 1] and the loop-invariant row-broadcast C = |p_m|^2.
  v2f a[4];
  v8f cc[4];
#pragma unroll
  for (int s = 0; s < 4; ++s) {
    const float4 w = Rb[rowBase + s * 16 + n16];
    a[s].x = hi ? (-0.5f * w.z) : (-0.5f * w.x);   // K2 : K0
    a[s].y = hi ? 1.0f          : (-0.5f * w.y);   // K3 : K1
    const float sq = w.w;                          // lane j holds |p_(row j)|^2
#pragma unroll
    for (int v = 0; v < 8; ++v)
      cc[s][v] = __shfl(sq, hi * 8 + v);           // C reg v = sq of row (hi*8+v)
  }

  v8f rowmin[4];
#pragma unroll
  for (int s = 0; s < 4; ++s)
#pragma unroll
    for (int v = 0; v < 8; ++v) rowmin[s][v] = 3.0e38f;

  // Two column tiles per step: both b128 loads issue before any consumer.
  const float4* cp = Cb + ccBase + n16;
  for (int ct = 0; ct < T; ct += 2) {
    const float4 w0 = cp[0];
    const float4 w1 = cp[16];
    cp += 32;

    v2f b0, b1;
    b0.x = hi ? w0.z : w0.x;                       // rows K2 : K0  (-2q)
    b0.y = hi ? w0.w : w0.y;                       // rows K3 : K1  (|q|^2 : -2qy)
    b1.x = hi ? w1.z : w1.x;
    b1.y = hi ? w1.w : w1.y;

#pragma unroll
    for (int s = 0; s < 4; ++s) {
      v8f d0 = __builtin_amdgcn_wmma_f32_16x16x4_f32(
          false, a[s], false, b0, (short)0, cc[s], false, false);
      v8f d1 = __builtin_amdgcn_wmma_f32_16x16x4_f32(
          false, a[s], false, b1, (short)0, cc[s], false, false);
#pragma unroll
      for (int v = 0; v < 8; ++v)
        rowmin[s][v] = min3_raw(rowmin[s][v], d0[v], d1[v]);  // 0.5 op/element
    }
  }

  // Min across the 16 lanes of each half (wave32: xor masks 1,2,4,8 stay in a
  // half); lane0 holds rows 0..7, lane16 rows 8..15 of each strip.
#pragma unroll
  for (int s = 0; s < 4; ++s) {
#pragma unroll
    for (int v = 0; v < 8; ++v) {
      float val = rowmin[s][v];
      val = min_raw(val, __shfl_xor(val, 1));
      val = min_raw(val, __shfl_xor(val, 2));
      val = min_raw(val, __shfl_xor(val, 4));
      val = min_raw(val, __shfl_xor(val, 8));
      rowmin[s][v] = val;
    }
    if (n16 == 0) {
      const int r0 = rowBase + s * 16 + hi * 8;
      float* drow = dpart + ((size_t)cs * 8 + batch) * NR;  // [cs][batch][row]
#pragma unroll
      for (int v = 0; v < 8; ++v)
        drow[r0 + v] = rowmin[s][v];               // raw partial min
    }
  }
}

// ---------------------------------------------------------------------------
// Per-batch reduction: min over CSPLIT partials, fold in the pad point, apply
// the deferred clamp (max(min(raw,·),0) == reference's clamp-then-min), mask
// zeros, and emit out[1+b] = sum1/pos1 + sum2/pos2. The reference's pad ROW
// has dist exactly 0 and is masked out, so it contributes nothing.
// ---------------------------------------------------------------------------
__global__ __launch_bounds__(256) void chamfer_reduce(
    const float* __restrict__ X, const float* __restrict__ Y,
    const float* __restrict__ d1part, const float* __restrict__ d2part,
    float* __restrict__ out, int N, int M)
{
  const int b = blockIdx.x;
  __shared__ float ss1[256], sc1[256], ss2[256], sc2[256];

  float sum1 = 0.0f, cnt1 = 0.0f, sum2 = 0.0f, cnt2 = 0.0f;
  const float* Xb = X + (size_t)b * N * 3;
  const float* Yb = Y + (size_t)b * M * 3;

  for (int i = threadIdx.x; i < N; i += 256) {
    float mn = d1part[((size_t)0 * 8 + b) * N + i];
#pragma unroll
    for (int cs = 1; cs < CSPLIT; ++cs)
      mn = fminf(mn, d1part[((size_t)cs * 8 + b) * N + i]);
    const float px = Xb[3 * i + 0] - PAD_VAL;
    const float py = Xb[3 * i + 1] - PAD_VAL;
    const float pz = Xb[3 * i + 2] - PAD_VAL;
    const float v = fmaxf(fminf(mn, px * px + py * py + pz * pz), 0.0f);
    if (v > 0.0f) { sum1 += v; cnt1 += 1.0f; }
  }
  for (int j = threadIdx.x; j < M; j += 256) {
    float mn = d2part[((size_t)0 * 8 + b) * M + j];
#pragma unroll
    for (int cs = 1; cs < CSPLIT; ++cs)
      mn = fminf(mn, d2part[((size_t)cs * 8 + b) * M + j]);
    const float qx = Yb[3 * j + 0] - PAD_VAL;
    const float qy = Yb[3 * j + 1] - PAD_VAL;
    const float qz = Yb[3 * j + 2] - PAD_VAL;
    const float v = fmaxf(fminf(mn, qx * qx + qy * qy + qz * qz), 0.0f);
    if (v > 0.0f) { sum2 += v; cnt2 += 1.0f; }
  }

  ss1[threadIdx.x] = sum1; sc1[threadIdx.x] = cnt1;
  ss2[threadIdx.x] = sum2; sc2[threadIdx.x] = cnt2;
  __syncthreads();
  for (int off = 128; off > 0; off >>= 1) {
    if (threadIdx.x < off) {
      ss1[threadIdx.x] += ss1[threadIdx.x + off];
      sc1[threadIdx.x] += sc1[threadIdx.x + off];
      ss2[threadIdx.x] += ss2[threadIdx.x + off];
      sc2[threadIdx.x] += sc2[threadIdx.x + off];
    }
    __syncthreads();
  }
  if (threadIdx.x == 0) {
    out[1 + b] = ss1[0] / fmaxf(sc1[0], 1.0f) + ss2[0] / fmaxf(sc2[0], 1.0f);
  }
}

// Scalar mean over the B per-batch values -> out[0].
__global__ void chamfer_mean(float* __restrict__ out, int B)
{
  if (threadIdx.x == 0 && blockIdx.x == 0) {
    float s = 0.0f;
    for (int b = 0; b < B; ++b) s += out[1 + b];
    out[0] = s / (float)B;
  }
}

extern "C" void kernel_launch(void* const* d_in, const int* in_sizes, int n_in,
                              void* d_out, int out_size, void* d_ws, size_t ws_size,
                              hipStream_t stream) {
  const float* X = (const float*)d_in[0];   // output_pc (B, N, 3) float32
  const float* Y = (const float*)d_in[1];   // target_pc (B, M, 3) float32
  const int B = 8;
  const int N = in_sizes[0] / (B * 3);      // 4096
  const int M = in_sizes[1] / (B * 3);      // 4096

  // Workspace: packed X | packed Y | d1 partials | d2 partials
  float4* XPw = (float4*)d_ws;
  float4* YPw = XPw + (size_t)B * N;
  float*  d1p = (float*)(YPw + (size_t)B * M);
  float*  d2p = d1p + (size_t)CSPLIT * B * N;
  float*  out = (float*)d_out;              // [mean, per-batch x 8]

  pack_points<<<(B * N + 255) / 256, 256, 0, stream>>>(X, XPw, B * N);
  pack_points<<<(B * M + 255) / 256, 256, 0, stream>>>(Y, YPw, B * M);

  // Both directions in one launch: blockIdx.y = direction.
  const int blocksX = (B * (N / 64) * CSPLIT) / 8;   // 8 waves per block
  nn_min_wmma<<<dim3(blocksX, 2), 256, 0, stream>>>(XPw, YPw, d1p, d2p, N, M);

  chamfer_reduce<<<B, 256, 0, stream>>>(X, Y, d1p, d2p, out, N, M);
  chamfer_mean<<<1, 32, 0, stream>>>(out, B);
}